// TensorProduct_39281770889328
// MI455X (gfx1250) — compile-verified
//
#include <hip/hip_runtime.h>

typedef float v2f __attribute__((ext_vector_type(2)));
typedef float v8f __attribute__((ext_vector_type(8)));

// ======================= compile-time Wigner-3j (e3nn convention) =====================
namespace cw {

constexpr double csqrt(double x) {
  if (x <= 0.0) return 0.0;
  double g = x > 1.0 ? x : 1.0;
  for (int i = 0; i < 64; ++i) g = 0.5 * (g + x / g);
  return g;
}
constexpr double dabs(double x) { return x < 0.0 ? -x : x; }
constexpr double fact(int n) { double r = 1.0; for (int i = 2; i <= n; ++i) r *= (double)i; return r; }
constexpr int imax3(int a, int b, int c) { int m = a > b ? a : b; return m > c ? m : c; }
constexpr int imin3(int a, int b, int c) { int m = a < b ? a : b; return m < c ? m : c; }

constexpr double su2_cg(int j1, int m1, int j2, int m2, int j3, int m3) {
  if (m3 != m1 + m2) return 0.0;
  int vmin = imax3(-j1 + j2 + m3, -j1 + m1, 0);
  int vmax = imin3(j2 + j3 + m1, j3 - j1 + j2, j3 + m3);
  double c = csqrt((2.0 * j3 + 1.0) * fact(j3 + j1 - j2) * fact(j3 - j1 + j2) * fact(j1 + j2 - j3) *
                   fact(j3 + m3) * fact(j3 - m3) /
                   (fact(j1 + j2 + j3 + 1) * fact(j1 - m1) * fact(j1 + m1) * fact(j2 - m2) * fact(j2 + m2)));
  double s = 0.0;
  for (int v = vmin; v <= vmax; ++v) {
    double sg = (((v + j2 + m2) & 1) == 0) ? 1.0 : -1.0;
    s += sg * fact(j2 + j3 + m1 - v) * fact(j1 - m1 + v) /
         (fact(v) * fact(j3 - j1 + j2 - v) * fact(j3 + m3 - v) * fact(v + j1 - j2 - m3));
  }
  return c * s;
}

struct Cplx { double re, im; };
constexpr Cplx cmul(Cplx a, Cplx b) { return Cplx{a.re * b.re - a.im * b.im, a.re * b.im + a.im * b.re}; }
constexpr Cplx cadd(Cplx a, Cplx b) { return Cplx{a.re + b.re, a.im + b.im}; }
constexpr Cplx cconj(Cplx a) { return Cplx{a.re, -a.im}; }

struct QMat { Cplx q[7][7]; };
constexpr QMat rtoc(int l) {   // real SH -> complex SH basis, times (-i)^l
  QMat Q{};
  double is2 = 1.0 / csqrt(2.0);
  for (int m = -l; m < 0; ++m) {
    Q.q[l + m][l - m] = Cplx{is2, 0.0};
    Q.q[l + m][l + m] = Cplx{0.0, -is2};
  }
  Q.q[l][l] = Cplx{1.0, 0.0};
  for (int m = 1; m <= l; ++m) {
    double s = ((m & 1) == 0) ? 1.0 : -1.0;
    Q.q[l + m][l + m] = Cplx{s * is2, 0.0};
    Q.q[l + m][l - m] = Cplx{0.0, s * is2};
  }
  Cplx ph{1.0, 0.0};
  for (int i = 0; i < l; ++i) ph = cmul(ph, Cplx{0.0, -1.0});
  for (int a = 0; a < 2 * l + 1; ++a)
    for (int b = 0; b < 2 * l + 1; ++b) Q.q[a][b] = cmul(ph, Q.q[a][b]);
  return Q;
}

struct W3J { double c[7][7][7]; };   // [2l1+1][2l2+1][2l3+1], includes *sqrt(2l3+1)
constexpr W3J wigner(int l1, int l2, int l3) {
  int n1 = 2 * l1 + 1, n2 = 2 * l2 + 1, n3 = 2 * l3 + 1;
  Cplx C[7][7][7]{};
  for (int m1 = -l1; m1 <= l1; ++m1)
    for (int m2 = -l2; m2 <= l2; ++m2) {
      int m3 = m1 + m2;
      if (m3 >= -l3 && m3 <= l3)
        C[l1 + m1][l2 + m2][l3 + m3] = Cplx{su2_cg(l1, m1, l2, m2, l3, m3), 0.0};
    }
  QMat Q1 = rtoc(l1), Q2 = rtoc(l2), Q3 = rtoc(l3);
  // einsum 'ij,kl,mn,ikn->jlm' with third operand conj(Q3^T): done as 3 single contractions
  Cplx T1[7][7][7]{};
  for (int j = 0; j < n1; ++j)
    for (int k = 0; k < n2; ++k)
      for (int n = 0; n < n3; ++n) {
        Cplx s{0.0, 0.0};
        for (int i = 0; i < n1; ++i) s = cadd(s, cmul(Q1.q[i][j], C[i][k][n]));
        T1[j][k][n] = s;
      }
  Cplx T2[7][7][7]{};
  for (int j = 0; j < n1; ++j)
    for (int l = 0; l < n2; ++l)
      for (int n = 0; n < n3; ++n) {
        Cplx s{0.0, 0.0};
        for (int k = 0; k < n2; ++k) s = cadd(s, cmul(Q2.q[k][l], T1[j][k][n]));
        T2[j][l][n] = s;
      }
  W3J R{};
  double nrm2 = 0.0;
  for (int j = 0; j < n1; ++j)
    for (int l = 0; l < n2; ++l)
      for (int m = 0; m < n3; ++m) {
        Cplx s{0.0, 0.0};
        for (int n = 0; n < n3; ++n) s = cadd(s, cmul(cconj(Q3.q[n][m]), T2[j][l][n]));
        R.c[j][l][m] = s.re;
        nrm2 += s.re * s.re;
      }
  double sc = csqrt((double)(2 * l3 + 1)) / csqrt(nrm2);
  for (int j = 0; j < n1; ++j)
    for (int l = 0; l < n2; ++l)
      for (int m = 0; m < n3; ++m) R.c[j][l][m] *= sc;
  return R;
}

// 17 paths, in output (io) order: l3=0 blocks, then l3=1, l3=2, l3=3.
constexpr W3J W00 = wigner(0, 0, 0);
constexpr W3J W01 = wigner(1, 1, 0);
constexpr W3J W02 = wigner(2, 2, 0);
constexpr W3J W03 = wigner(0, 1, 1);
constexpr W3J W04 = wigner(1, 0, 1);
constexpr W3J W05 = wigner(1, 2, 1);
constexpr W3J W06 = wigner(2, 1, 1);
constexpr W3J W07 = wigner(2, 3, 1);
constexpr W3J W08 = wigner(0, 2, 2);
constexpr W3J W09 = wigner(1, 1, 2);
constexpr W3J W10 = wigner(1, 3, 2);
constexpr W3J W11 = wigner(2, 0, 2);
constexpr W3J W12 = wigner(2, 2, 2);
constexpr W3J W13 = wigner(0, 3, 3);
constexpr W3J W14 = wigner(1, 2, 3);
constexpr W3J W15 = wigner(2, 1, 3);
constexpr W3J W16 = wigner(2, 3, 3);
constexpr W3J CGS[17] = {W00, W01, W02, W03, W04, W05, W06, W07, W08,
                         W09, W10, W11, W12, W13, W14, W15, W16};

// convention sanity checks: (0,l,l) and (l,0,l) scaled couplings are identity deltas
static_assert(dabs(CGS[0].c[0][0][0] - 1.0) < 1e-9, "w3j(0,0,0) convention");
static_assert(dabs(CGS[3].c[0][1][1] - 1.0) < 1e-9, "w3j(0,1,1) convention");
static_assert(dabs(CGS[4].c[1][0][1] - 1.0) < 1e-9, "w3j(1,0,1) convention");

constexpr int MAXCELLS = 256;
constexpr int MAXTERMS = 1152;

struct TPTables {
  int ncells, nterms, totCols, totOut;
  short cellRow[MAXCELLS];        // row in 9-row x1 space (0..8)
  short cellCol[MAXCELLS];        // column in 71-col T space
  short cellStart[MAXCELLS + 1];  // term ranges
  short termJ[MAXTERMS];          // x2 index 0..15
  float termC[MAXTERMS];          // CG coefficient
  int outBase[17];                // output base per block io
  int colBase[17];                // T column base per block io
  int d3[17];                     // 2*l3+1
  int umult[17];                  // magic for /d3 (16-bit reciprocal, exact for rel<896)
};

constexpr TPTables build_tables() {
  TPTables t{};
  const int i1s[17] = {0, 1, 2, 0, 1, 1, 2, 2, 0, 1, 1, 2, 2, 0, 1, 2, 2};  // also l1
  const int i2s[17] = {0, 1, 2, 1, 0, 2, 1, 3, 2, 1, 3, 0, 2, 3, 2, 1, 3};  // also l2
  const int l3s[17] = {0, 0, 0, 1, 1, 1, 1, 1, 2, 2, 2, 2, 2, 3, 3, 3, 3};
  const int rowOff[3] = {0, 1, 4};       // x1 row offset per i1 (l=0,1,2)
  const int jOff[4] = {0, 1, 4, 9};      // x2 offset per i2 (l=0..3)
  int colbase = 0, outbase = 0, nc = 0, nt = 0;
  t.cellStart[0] = 0;
  for (int io = 0; io < 17; ++io) {
    int l1 = i1s[io], l2 = i2s[io], l3 = l3s[io];
    int d3 = 2 * l3 + 1;
    t.outBase[io] = outbase;
    t.colBase[io] = colbase;
    t.d3[io] = d3;
    t.umult[io] = (65536 + d3 - 1) / d3;
    for (int i = 0; i < 2 * l1 + 1; ++i)
      for (int k = 0; k < d3; ++k) {
        int st = nt;
        for (int j = 0; j < 2 * l2 + 1; ++j) {
          double v = CGS[io].c[i][j][k];
          if (v > 1e-12 || v < -1e-12) {
            t.termJ[nt] = (short)(jOff[i2s[io]] + j);
            t.termC[nt] = (float)v;
            ++nt;
          }
        }
        if (nt > st) {
          t.cellRow[nc] = (short)(rowOff[i1s[io]] + i);
          t.cellCol[nc] = (short)(colbase + k);
          ++nc;
          t.cellStart[nc] = (short)nt;
        }
      }
    colbase += d3;
    outbase += 128 * d3;
  }
  t.ncells = nc;
  t.nterms = nt;
  t.totCols = colbase;
  t.totOut = outbase;
  return t;
}

constexpr TPTables TAB = build_tables();
static_assert(TAB.totCols == 71, "column layout mismatch");
static_assert(TAB.totOut == 9088, "output layout mismatch");
static_assert(TAB.ncells <= MAXCELLS, "cell overflow");
static_assert(TAB.nterms <= MAXTERMS, "term overflow");

}  // namespace cw

__constant__ cw::TPTables d_tab = cw::TAB;

// =============================== kernel ===============================
// One 256-thread (8-wave, wave32) block per batch.
//   T[128 x 71] = x1row[128 x 9(pad12)] @ Wmat[9(pad12) x 71(pad80)]
// via v_wmma_f32_16x16x4_f32:  8 M-tiles (1 per wave) x 5 N-tiles x 3 K-steps.
__global__ __launch_bounds__(256) void tp_wmma_kernel(const float* __restrict__ x1,
                                                      const float* __restrict__ x2,
                                                      const float* __restrict__ wt,
                                                      float* __restrict__ out) {
  __shared__ float s_x2[16];
  __shared__ float s_x1[128 * 12];  // padded K=12, [u][i]
  __shared__ float s_wm[12 * 80];   // Wmat, padded 12 x 80, [i][col]
  __shared__ float s_w[2176];       // weights for this batch
  __shared__ float s_t[128 * 81];   // GEMM result T, stride 81 (bank-conflict pad)

  const int b = blockIdx.x;
  const int t = threadIdx.x;
  const float* x1b = x1 + (size_t)b * 1152;
  const float* x2b = x2 + (size_t)b * 16;
  const float* wb = wt + (size_t)b * 2176;
  float* ob = out + (size_t)b * 9088;

  // Phase A: zero padded buffers, stage x2
  if (t < 16) s_x2[t] = x2b[t];
  for (int g = t; g < 128 * 12; g += 256) s_x1[g] = 0.0f;
  for (int g = t; g < 12 * 80; g += 256) s_wm[g] = 0.0f;
  __syncthreads();

  // Phase B: coalesced x1 -> padded [u][i] layout; weights -> LDS; Wmat cells from x2
  for (int g = t; g < 1152; g += 256) {
    int u, i;
    if (g < 128) { u = g; i = 0; }
    else if (g < 512) { int q = g - 128; u = q / 3; i = 1 + (q - u * 3); }
    else { int q = g - 512; u = q / 5; i = 4 + (q - u * 5); }
    s_x1[u * 12 + i] = x1b[g];
  }
  for (int g = t; g < 2176; g += 256) s_w[g] = wb[g];
  {
    const int nc = d_tab.ncells;
    for (int c0 = t; c0 < nc; c0 += 256) {
      const int s0 = d_tab.cellStart[c0], s1 = d_tab.cellStart[c0 + 1];
      float a = 0.0f;
      for (int s = s0; s < s1; ++s) a += d_tab.termC[s] * s_x2[d_tab.termJ[s]];
      s_wm[d_tab.cellRow[c0] * 80 + d_tab.cellCol[c0]] = a;
    }
  }
  __syncthreads();

  // Phase C: WMMA GEMM. Wave w owns M-rows u = 16w..16w+15.
  const int w = t >> 5;
  const int lane = t & 31;
  const int lm = lane & 15;
  const int kh = (lane < 16) ? 0 : 2;  // K half-select per documented 32-bit A/B layouts
  const float* xr = &s_x1[(16 * w + lm) * 12];
  v2f a0 = {xr[kh + 0], xr[kh + 1]};   // K-step 0: K = 0..3
  v2f a1 = {xr[kh + 4], xr[kh + 5]};   // K-step 1: K = 4..7
  v2f a2 = {xr[kh + 8], xr[kh + 9]};   // K-step 2: K = 8..11 (9..11 are zero pad)
  const int mb = 16 * w + ((lane < 16) ? 0 : 8);
#pragma unroll
  for (int n = 0; n < 5; ++n) {
    const int nc0 = 16 * n + lm;
    v2f b0 = {s_wm[(kh + 0) * 80 + nc0], s_wm[(kh + 1) * 80 + nc0]};
    v2f b1 = {s_wm[(kh + 4) * 80 + nc0], s_wm[(kh + 5) * 80 + nc0]};
    v2f b2 = {s_wm[(kh + 8) * 80 + nc0], s_wm[(kh + 9) * 80 + nc0]};
    v8f acc = {};
    acc = __builtin_amdgcn_wmma_f32_16x16x4_f32(false, a0, false, b0, (short)0, acc, false, false);
    acc = __builtin_amdgcn_wmma_f32_16x16x4_f32(false, a1, false, b1, (short)0, acc, false, false);
    acc = __builtin_amdgcn_wmma_f32_16x16x4_f32(false, a2, false, b2, (short)0, acc, false, false);
#pragma unroll
    for (int r = 0; r < 8; ++r) s_t[(mb + r) * 81 + nc0] = acc[r];
  }
  __syncthreads();

  // Phase D: apply per-(u,instruction) weight, fully coalesced global store
  for (int idx = t; idx < 9088; idx += 256) {
    int io = 0;
#pragma unroll
    for (int q = 1; q < 17; ++q) io = (idx >= d_tab.outBase[q]) ? q : io;
    const int rel = idx - d_tab.outBase[io];
    const int d3 = d_tab.d3[io];
    const int u = (int)(((unsigned)rel * (unsigned)d_tab.umult[io]) >> 16);
    const int k = rel - u * d3;
    ob[idx] = s_w[io * 128 + u] * s_t[u * 81 + d_tab.colBase[io] + k];
  }
}

extern "C" void kernel_launch(void* const* d_in, const int* in_sizes, int n_in,
                              void* d_out, int out_size, void* d_ws, size_t ws_size,
                              hipStream_t stream) {
  (void)n_in; (void)out_size; (void)d_ws; (void)ws_size;
  const float* x1 = (const float*)d_in[0];
  const float* x2 = (const float*)d_in[1];
  const float* wt = (const float*)d_in[2];
  float* out = (float*)d_out;
  const int B = in_sizes[0] / 1152;
  if (B <= 0) return;
  tp_wmma_kernel<<<dim3(B), dim3(256), 0, stream>>>(x1, x2, wt, out);
}